// Loop_Projection_42090679501535
// MI455X (gfx1250) — compile-verified
//
#include <hip/hip_runtime.h>

// Loop projection: out[b,e,p] = sum_d x[b,d,p] * W[p,d,e] + bias[p,e]
// B=256, D=512, P=128, E=128, fp32.
//
// Per-prototype GEMMs on V_WMMA_F32_16X16X4_F32. 8 prototypes per workgroup
// (one per wave) so global x/out access gets contiguous p-runs; LDS-transposed
// staging so each WMMA fragment is one conflict-free ds_load_b64 (pairs fused
// by the compiler into ds_load_2addr_b64). Double-buffered LDS: next chunk's
// global loads are issued before compute, consumed into LDS after compute,
// one barrier per chunk.
//
// Roofline: 4.29 GFLOP, working set 112 MiB (fits 192 MB L2). HBM floor ~5us;
// fp32 matrix throughput is the binding constraint, so maximize WMMA issue
// density: 4 e-tiles per wave => 16 WMMAs per chunk against ~11 LDS frag ops.

typedef __attribute__((ext_vector_type(2))) float v2f;
typedef __attribute__((ext_vector_type(4))) float v4f;
typedef __attribute__((ext_vector_type(8))) float v8f;

#define B_SZ 256
#define D_SZ 512
#define P_SZ 128
#define E_SZ 128
#define XB_PITCH (D_SZ * P_SZ)   // 65536: x batch stride
#define WP_PITCH (D_SZ * E_SZ)   // 65536: W prototype stride
#define OB_PITCH (E_SZ * P_SZ)   // 16384: out batch stride

#define KC     16                // d-chunk per LDS stage (4 WMMA k-steps)
#define KPAD   18                // KC+2: even (b64 align), conflict-free stride
#define PB      8                // prototypes per workgroup (one per wave)
#define BTILE  16                // batch rows per workgroup
#define ETILE  64                // e cols per workgroup = 4 WMMA tiles per wave
#define NCHUNK (D_SZ / KC)       // 32

__global__ __launch_bounds__(256, 2)
void loop_proj_wmma_kernel(const float* __restrict__ x,
                           const float* __restrict__ W,
                           const float* __restrict__ bias,
                           float* __restrict__ out)
{
    __shared__ float sX[2][PB * BTILE * KPAD];   // [pi][b][k]  2 x  9216 B
    __shared__ float sW[2][PB * ETILE * KPAD];   // [pi][e][k]  2 x 36864 B

    const int tid  = threadIdx.x;
    const int lane = tid & 31;
    const int wave = tid >> 5;                   // = pi of this wave's prototype

    const int pg0 = blockIdx.x * PB;             // prototype group base
    const int b0  = blockIdx.y * BTILE;          // batch tile base
    const int e0  = blockIdx.z * ETILE;          // e tile base

    // WMMA 16x16x4 f32 fragment decomposition:
    //   lanes 0-15:  M/N = lane,    VGPR0/1 = K,K+1
    //   lanes 16-31: M/N = lane-16, VGPR0/1 = K+2,K+3
    const int m    = lane & 15;
    const int koff = (lane >> 4) << 1;           // 0 or 2

    v8f acc[4] = {};                             // e sub-tiles: e0+0/16/32/48

    // ---- x staging map: thread pair covers one (i,k); t&1 picks p float4 half.
    //      rep 0 -> rows i in [0,8), rep 1 -> i+8 (same k).
    const int pairX = tid >> 1;
    const int phalf = (tid & 1) * 4;
    const int iX    = pairX >> 4;                // 0..7
    const int kX    = pairX & 15;
    const float* xp0 = x + (size_t)(b0 + iX) * XB_PITCH
                         + (size_t)kX * P_SZ + pg0 + phalf;
    const float* xp1 = xp0 + (size_t)8 * XB_PITCH;

    // ---- W staging map: thread pair covers one (pi,k); t&1 picks e half of 32.
    const int piW   = (tid >> 1) >> 4;           // 0..7
    const int kW    = (tid >> 1) & 15;           // 0..15
    const int eHalf = (tid & 1) * 32;
    const float* wp = W + (size_t)(pg0 + piW) * WP_PITCH
                        + (size_t)kW * E_SZ + e0 + eHalf;

    // LDS store helpers (chunk-invariant addresses)
    auto stage = [&](int buf, const v4f& rx0, const v4f& rx1, const v4f (&rw)[8]) {
        #pragma unroll
        for (int j = 0; j < 4; ++j) {
            sX[buf][((phalf + j) * BTILE + iX)     * KPAD + kX] = rx0[j];
            sX[buf][((phalf + j) * BTILE + iX + 8) * KPAD + kX] = rx1[j];
        }
        #pragma unroll
        for (int j = 0; j < 8; ++j) {
            const int e = eHalf + j * 4;
            #pragma unroll
            for (int q = 0; q < 4; ++q)
                sW[buf][(piW * ETILE + e + q) * KPAD + kW] = rw[j][q];
        }
    };

    // ---- prologue: chunk 0 -> buf 0
    {
        v4f rx0 = *(const v4f*)xp0;
        v4f rx1 = *(const v4f*)xp1;
        v4f rw[8];
        #pragma unroll
        for (int j = 0; j < 8; ++j) rw[j] = *(const v4f*)(wp + j * 4);
        stage(0, rx0, rx1, rw);
    }

    for (int c = 0; c < NCHUNK; ++c) {
        const int cur  = c & 1;
        const bool more = (c + 1) < NCHUNK;

        // issue next chunk's global loads before compute (latency overlap)
        v4f nrx0, nrx1, nrw[8];
        if (more) {
            const size_t off = (size_t)(c + 1) * KC * P_SZ;  // both pitches = 128/d
            nrx0 = *(const v4f*)(xp0 + off);
            nrx1 = *(const v4f*)(xp1 + off);
            #pragma unroll
            for (int j = 0; j < 8; ++j) nrw[j] = *(const v4f*)(wp + off + j * 4);
        }

        // barrier: buf[cur] stores visible; all waves done reading buf[cur^1]
        __syncthreads();

        // ---- 4 WMMA k-steps x 4 e-tiles over this chunk
        const float* bX = sX[cur];
        const float* bW = sW[cur];
        #pragma unroll
        for (int k0 = 0; k0 < KC; k0 += 4) {
            v2f a = *(const v2f*)&bX[(wave * BTILE + m) * KPAD + k0 + koff];
            #pragma unroll
            for (int t = 0; t < 4; ++t) {
                v2f bf = *(const v2f*)&bW[(wave * ETILE + t * 16 + m) * KPAD + k0 + koff];
                // 8 args: (neg_a, A, neg_b, B, c_mod, C, reuse_a, reuse_b)
                acc[t] = __builtin_amdgcn_wmma_f32_16x16x4_f32(
                             false, a, false, bf, (short)0, acc[t], false, false);
            }
        }

        // consume staged regs into the other buffer (safe: post-barrier)
        if (more) stage(cur ^ 1, nrx0, nrx1, nrw);
    }

    // ---- epilogue: bias add + store.
    // C/D layout: VGPR r -> M=r (lanes 0-15) or M=r+8 (lanes 16-31); N = lane&15.
    const int p    = pg0 + wave;
    const int mtop = (lane >> 4) << 3;           // 0 or 8
    float bi[4];
    #pragma unroll
    for (int t = 0; t < 4; ++t) bi[t] = bias[p * E_SZ + e0 + t * 16 + m];

    #pragma unroll
    for (int r = 0; r < 8; ++r) {
        const size_t bb = (size_t)(b0 + r + mtop) * OB_PITCH;
        #pragma unroll
        for (int t = 0; t < 4; ++t)
            out[bb + (size_t)(e0 + t * 16 + m) * P_SZ + p] = acc[t][r] + bi[t];
    }
}

extern "C" void kernel_launch(void* const* d_in, const int* in_sizes, int n_in,
                              void* d_out, int out_size, void* d_ws, size_t ws_size,
                              hipStream_t stream) {
    (void)in_sizes; (void)n_in; (void)d_ws; (void)ws_size; (void)out_size;
    const float* x    = (const float*)d_in[0];   // [B,D,P]
    const float* W    = (const float*)d_in[1];   // [P,D,E]
    const float* bias = (const float*)d_in[2];   // [P,E]
    float* out        = (float*)d_out;           // [B,E,P]

    dim3 grid(P_SZ / PB, B_SZ / BTILE, E_SZ / ETILE);  // (16, 16, 2)
    dim3 block(256);
    loop_proj_wmma_kernel<<<grid, block, 0, stream>>>(x, W, bias, out);
}